// PointCNN_76982993813719
// MI455X (gfx1250) — compile-verified
//
#include <hip/hip_runtime.h>
#include <math.h>

// PointCNN forward for MI455X (gfx1250). All GEMMs use V_WMMA_F32_16X16X4_F32
// (fp32 keeps reference numerics; network is latency-bound, not FLOP-bound,
// so low-precision WMMA would only add conversion traffic).

#define N_B   8
#define NPTSC 2048

typedef __attribute__((ext_vector_type(2))) float v2f;
typedef __attribute__((ext_vector_type(8))) float v8f;

#define ACT_NONE 0
#define ACT_ELU  1
#define ACT_SIG  2

__device__ __forceinline__ float act_f(float v, int act) {
  if (act == ACT_ELU) return v > 0.f ? v : (__expf(v) - 1.f);
  if (act == ACT_SIG) return 1.f / (1.f + __expf(-v));
  return v;
}

// ---------------------------------------------------------------------------
// Generic dense GEMM: D[m, coff+n] = act(sum_k A[m,k]*W[k,n] + bias[n])
// One wave computes a 16x32 slab (two 16x16 WMMA tiles sharing the A frag).
// Bounds: row/col are CLAMPED (garbage only lands in store-masked outputs);
// k is exact: unguarded multiple-of-8 loop + one guarded remainder step.
// A-matrix lane layout (16x4 f32): lanes 0-15 hold K=0,1; lanes 16-31 K=2,3.
// B-matrix (4x16): lanes 0-15 rows K=0,1; lanes 16-31 rows K=2,3.
// C/D (16x16): VGPR r -> row tm+r (lanes 0-15) / row tm+r+8 (lanes 16-31).
// ---------------------------------------------------------------------------
__global__ __launch_bounds__(32) void dense_wmma_k(
    const float* __restrict__ A, int lda, int rows, int cin,
    const float* __restrict__ W, const float* __restrict__ bias,
    float* __restrict__ D, int ldd, int coff, int cout, int act)
{
  const int tm   = blockIdx.x * 16;
  const int tn   = blockIdx.y * 32;
  const int lane = threadIdx.x & 31;
  const int half = lane >> 4;
  const int l15  = lane & 15;
  const int row  = tm + l15;
  const int col0 = tn + l15;
  const int col1 = tn + 16 + l15;

  const int rowc  = row  < rows ? row  : rows - 1;
  const int col0c = col0 < cout ? col0 : cout - 1;
  const int col1c = col1 < cout ? col1 : cout - 1;

  const float* __restrict__ Ap = A + (size_t)rowc * lda;
  const float* __restrict__ W0 = W + col0c;
  const float* __restrict__ W1 = W + col1c;

  v8f acc0 = {0.f, 0.f, 0.f, 0.f, 0.f, 0.f, 0.f, 0.f};
  v8f acc1 = {0.f, 0.f, 0.f, 0.f, 0.f, 0.f, 0.f, 0.f};

  const int kfull = cin & ~7;
  int k0 = 0;
  for (; k0 < kfull; k0 += 8) {
    const int ka = k0 + half * 2;
    v2f a0, a1, b00, b10, b01, b11;
    a0.x  = Ap[ka];     a0.y  = Ap[ka + 1];
    a1.x  = Ap[ka + 4]; a1.y  = Ap[ka + 5];
    b00.x = W0[(size_t)(ka    ) * cout]; b00.y = W0[(size_t)(ka + 1) * cout];
    b10.x = W1[(size_t)(ka    ) * cout]; b10.y = W1[(size_t)(ka + 1) * cout];
    b01.x = W0[(size_t)(ka + 4) * cout]; b01.y = W0[(size_t)(ka + 5) * cout];
    b11.x = W1[(size_t)(ka + 4) * cout]; b11.y = W1[(size_t)(ka + 5) * cout];
    __builtin_prefetch(Ap + k0 + 8, 0, 1);  // global_prefetch_b8
    acc0 = __builtin_amdgcn_wmma_f32_16x16x4_f32(false, a0, false, b00, (short)0, acc0, false, false);
    acc1 = __builtin_amdgcn_wmma_f32_16x16x4_f32(false, a0, false, b10, (short)0, acc1, false, false);
    acc0 = __builtin_amdgcn_wmma_f32_16x16x4_f32(false, a1, false, b01, (short)0, acc0, false, false);
    acc1 = __builtin_amdgcn_wmma_f32_16x16x4_f32(false, a1, false, b11, (short)0, acc1, false, false);
  }
  for (; k0 < cin; k0 += 4) {  // k remainder (guarded; at most one trip)
    const int ka = k0 + half * 2;
    v2f a, b0, b1;
    a.x  = (ka     < cin) ? Ap[ka]     : 0.f;
    a.y  = (ka + 1 < cin) ? Ap[ka + 1] : 0.f;
    b0.x = (ka     < cin) ? W0[(size_t)(ka    ) * cout] : 0.f;
    b0.y = (ka + 1 < cin) ? W0[(size_t)(ka + 1) * cout] : 0.f;
    b1.x = (ka     < cin) ? W1[(size_t)(ka    ) * cout] : 0.f;
    b1.y = (ka + 1 < cin) ? W1[(size_t)(ka + 1) * cout] : 0.f;
    acc0 = __builtin_amdgcn_wmma_f32_16x16x4_f32(false, a, false, b0, (short)0, acc0, false, false);
    acc1 = __builtin_amdgcn_wmma_f32_16x16x4_f32(false, a, false, b1, (short)0, acc1, false, false);
  }

  if (col0 < cout) {
    const float bv = bias[col0];
#pragma unroll
    for (int r = 0; r < 8; ++r) {
      const int m = tm + r + half * 8;
      if (m < rows) D[(size_t)m * ldd + coff + col0] = act_f(acc0[r] + bv, act);
    }
  }
  if (col1 < cout) {
    const float bv = bias[col1];
#pragma unroll
    for (int r = 0; r < 8; ++r) {
      const int m = tm + r + half * 8;
      if (m < rows) D[(size_t)m * ldd + coff + col1] = act_f(acc1[r] + bv, act);
    }
  }
}

// ---------------------------------------------------------------------------
// Brute-force KNN: one thread per query keeps a sorted top-KSEL list,
// then emits every D-th neighbor (K of them).
// ---------------------------------------------------------------------------
template <int KSEL, int DD, int KK>
__global__ __launch_bounds__(128) void knn_k(
    const float* __restrict__ qrs, int P, const float* __restrict__ pts, int M,
    int* __restrict__ idx)
{
  const int g = blockIdx.x * blockDim.x + threadIdx.x;
  if (g >= N_B * P) return;
  const int n = g / P;
  const float qx = qrs[(size_t)g * 3 + 0];
  const float qy = qrs[(size_t)g * 3 + 1];
  const float qz = qrs[(size_t)g * 3 + 2];
  float bd[KSEL];
  int   bi[KSEL];
#pragma unroll
  for (int i = 0; i < KSEL; ++i) { bd[i] = 3.4e38f; bi[i] = 0; }
  const float* bp = pts + (size_t)n * M * 3;
  for (int m = 0; m < M; ++m) {
    const float dx = bp[3 * m + 0] - qx;
    const float dy = bp[3 * m + 1] - qy;
    const float dz = bp[3 * m + 2] - qz;
    const float d  = dx * dx + dy * dy + dz * dz;
    if (d < bd[KSEL - 1]) {
      int j = KSEL - 1;
      while (j > 0 && bd[j - 1] > d) { bd[j] = bd[j - 1]; bi[j] = bi[j - 1]; --j; }
      bd[j] = d; bi[j] = m;
    }
  }
#pragma unroll
  for (int k = 0; k < KK; ++k) idx[(size_t)g * KK + k] = bi[k * DD];
}

// ---------------------------------------------------------------------------
// Farthest point sampling: one block per batch; dist[] lives in LDS.
// Matches lax.scan semantics: emit `last` before the distance update.
// ---------------------------------------------------------------------------
__global__ __launch_bounds__(256) void fps_k(
    const float* __restrict__ pts, int M, int P, int* __restrict__ outidx)
{
  __shared__ float dist[NPTSC];
  __shared__ float rv[256];
  __shared__ int   ri[256];
  __shared__ int   lastS;
  const int n = blockIdx.x;
  const int t = threadIdx.x;
  if (t == 0) lastS = 0;
  for (int i = t; i < M; i += 256) dist[i] = 3.4e38f;
  __syncthreads();
  const float* bp = pts + (size_t)n * M * 3;
  for (int it = 0; it < P; ++it) {
    const int L = lastS;
    if (t == 0) outidx[n * P + it] = L;
    const float lx = bp[3 * L + 0], ly = bp[3 * L + 1], lz = bp[3 * L + 2];
    float bv = -1.f; int bj = 0;
    for (int i = t; i < M; i += 256) {
      const float dx = bp[3 * i + 0] - lx;
      const float dy = bp[3 * i + 1] - ly;
      const float dz = bp[3 * i + 2] - lz;
      const float nd = fminf(dist[i], dx * dx + dy * dy + dz * dz);
      dist[i] = nd;
      if (nd > bv) { bv = nd; bj = i; }
    }
    rv[t] = bv; ri[t] = bj;
    __syncthreads();
    for (int s = 128; s > 0; s >>= 1) {
      if (t < s) {
        if (rv[t + s] > rv[t] || (rv[t + s] == rv[t] && ri[t + s] < ri[t])) {
          rv[t] = rv[t + s]; ri[t] = ri[t + s];
        }
      }
      __syncthreads();
    }
    if (t == 0) lastS = ri[0];
    __syncthreads();
  }
}

// qrs[g] = pts[n, fidx[g]]
__global__ void gather3_k(const float* __restrict__ pts, const int* __restrict__ fidx,
                          int M, int P, float* __restrict__ q)
{
  const int g = blockIdx.x * blockDim.x + threadIdx.x;
  if (g >= N_B * P) return;
  const int n = g / P;
  const int s = fidx[g];
  q[(size_t)g * 3 + 0] = pts[((size_t)n * M + s) * 3 + 0];
  q[(size_t)g * 3 + 1] = pts[((size_t)n * M + s) * 3 + 1];
  q[(size_t)g * 3 + 2] = pts[((size_t)n * M + s) * 3 + 2];
}

// loc[n,p,k,:] = pts[n, idx[n,p,k], :] - qrs[n,p,:]
__global__ void gather_loc_k(const float* __restrict__ pts, const float* __restrict__ qrs,
                             const int* __restrict__ idx, int K, int P, int M,
                             float* __restrict__ loc)
{
  const int g = blockIdx.x * blockDim.x + threadIdx.x;
  if (g >= N_B * P * K) return;
  const int p = (g / K) % P;
  const int n = g / (K * P);
  const int s = idx[g];
  const float* sp = &pts[((size_t)n * M + s) * 3];
  const float* qp = &qrs[((size_t)n * P + p) * 3];
  float* lp = &loc[(size_t)g * 3];
  lp[0] = sp[0] - qp[0];
  lp[1] = sp[1] - qp[1];
  lp[2] = sp[2] - qp[2];
}

// generic strided column copy (pts0 extraction, fuse concat)
__global__ void copy_cols_k(const float* __restrict__ src, int lds, int soff,
                            float* __restrict__ dst, int ldd, int doff,
                            int rows, int cols)
{
  const int g = blockIdx.x * blockDim.x + threadIdx.x;
  if (g >= rows * cols) return;
  const int r = g / cols, c = g % cols;
  dst[(size_t)r * ldd + doff + c] = src[(size_t)r * lds + soff + c];
}

// out[n,p,c,m] = act(sum_k in[n,p,k,c] * w[k,c,m] + b[c,m]);  one block/point
template <int KK>
__global__ __launch_bounds__(KK * KK) void ein_k(
    const float* __restrict__ in, const float* __restrict__ w,
    const float* __restrict__ b, float* __restrict__ out, int act)
{
  __shared__ float xs[KK * KK];
  const int p = blockIdx.x;
  const int t = threadIdx.x;
  xs[t] = in[(size_t)p * KK * KK + t];
  __syncthreads();
  const int c = t / KK, m = t % KK;
  float s = b[t];
#pragma unroll
  for (int k = 0; k < KK; ++k) s += xs[k * KK + c] * w[((size_t)k * KK + c) * KK + m];
  out[(size_t)p * KK * KK + t] = act_f(s, act);
}

// Fused fX + depthwise: avoids materializing nn_fts / fX.
//   F[j,c]  = (c<cpf) ? h1[(bp*K+j)*cpf+c] : fts[n, idx[bp*K+j], c-cpf]
//   fx[i,c] = sum_j x2[bp,i,j] * F[j,c]
//   out[bp, c*dm+m] = sum_k fx[k,c] * dww[k,c,m]
template <int KK>
__global__ __launch_bounds__(256) void fx_dw_k(
    const float* __restrict__ x2, const float* __restrict__ h1,
    const float* __restrict__ fts, const int* __restrict__ idx,
    int cpf, int Cf, int cin, int dm, int M, int P,
    const float* __restrict__ dww, float* __restrict__ out)
{
  __shared__ float xs[KK * KK];
  __shared__ int   sidx[KK];
  const int bp = blockIdx.x;
  const int n  = bp / P;
  const int t  = threadIdx.x;
  if (t < KK * KK) xs[t] = x2[(size_t)bp * KK * KK + t];
  if (t < KK)      sidx[t] = idx[(size_t)bp * KK + t];
  __syncthreads();
  for (int c = t; c < cin; c += blockDim.x) {
    float fcol[KK];
#pragma unroll
    for (int j = 0; j < KK; ++j) {
      fcol[j] = (c < cpf) ? h1[((size_t)bp * KK + j) * cpf + c]
                          : fts[((size_t)n * M + sidx[j]) * Cf + (c - cpf)];
    }
    float fx[KK];
#pragma unroll
    for (int i = 0; i < KK; ++i) {
      float s = 0.f;
#pragma unroll
      for (int j = 0; j < KK; ++j) s += xs[i * KK + j] * fcol[j];
      fx[i] = s;
    }
    for (int m = 0; m < dm; ++m) {
      float s = 0.f;
#pragma unroll
      for (int k = 0; k < KK; ++k) s += fx[k] * dww[((size_t)k * cin + c) * dm + m];
      out[(size_t)bp * cin * dm + (size_t)c * dm + m] = s;
    }
  }
}

// ---------------------------------------------------------------------------
// Host-side orchestration
// ---------------------------------------------------------------------------
struct XcP {
  const float *dww, *f0b, *f0w, *f1b, *f1w, *pwb, *pww,
              *x0b, *x0w, *x1b, *x1w, *x2b, *x2w;
};

static XcP mk_xc(void* const* din, int b, bool wg) {
  XcP p;
  const int o = wg ? 4 : 0;  // skip g0{b,w}, g1{b,w} (alphabetical after fts1)
  p.dww = (const float*)din[b + 0];
  p.f0b = (const float*)din[b + 1];  p.f0w = (const float*)din[b + 2];
  p.f1b = (const float*)din[b + 3];  p.f1w = (const float*)din[b + 4];
  p.pwb = (const float*)din[b + 5 + o];  p.pww = (const float*)din[b + 6 + o];
  p.x0b = (const float*)din[b + 7 + o];  p.x0w = (const float*)din[b + 8 + o];
  p.x1b = (const float*)din[b + 9 + o];  p.x1w = (const float*)din[b + 10 + o];
  p.x2b = (const float*)din[b + 11 + o]; p.x2w = (const float*)din[b + 12 + o];
  return p;
}

static void dense(hipStream_t st, const float* A, int lda, int rows, int cin,
                  const float* W, const float* B, float* D, int ldd, int coff,
                  int cout, int act)
{
  dim3 grid((unsigned)((rows + 15) / 16), (unsigned)((cout + 31) / 32));
  dense_wmma_k<<<grid, 32, 0, st>>>(A, lda, rows, cin, W, B, D, ldd, coff, cout, act);
}

static void knn(hipStream_t st, const float* qrs, int P, const float* pts, int M,
                int K, int D, int* idx)
{
  dim3 g((unsigned)((N_B * P + 127) / 128));
  if (K == 12 && D == 1)      knn_k<12, 1, 12><<<g, 128, 0, st>>>(qrs, P, pts, M, idx);
  else if (K == 16 && D == 1) knn_k<16, 1, 16><<<g, 128, 0, st>>>(qrs, P, pts, M, idx);
  else                        knn_k<32, 2, 16><<<g, 128, 0, st>>>(qrs, P, pts, M, idx);
}

static void run_xconv(hipStream_t st, const XcP& W,
                      const float* pts, int M, const float* fts, int Cf,
                      const float* qrs, int Pq, int K, int D, int cpf, int dm, int C,
                      float* out, int ldd, int coff,
                      int* idxb, float* locb, float* h0, float* h1,
                      float* xa, float* xb, float* dwo)
{
  const int NP  = N_B * Pq;
  const int cin = cpf + Cf;
  knn(st, qrs, Pq, pts, M, K, D, idxb);
  {
    const int tot = NP * K;
    gather_loc_k<<<(tot + 255) / 256, 256, 0, st>>>(pts, qrs, idxb, K, Pq, M, locb);
  }
  dense(st, locb, 3,     NP * K, 3,     W.f0w, W.f0b, h0, cpf,   0, cpf,   ACT_ELU);
  dense(st, h0,   cpf,   NP * K, cpf,   W.f1w, W.f1b, h1, cpf,   0, cpf,   ACT_ELU);
  dense(st, locb, 3 * K, NP,     3 * K, W.x0w, W.x0b, xa, K * K, 0, K * K, ACT_ELU);
  if (K == 12) {
    ein_k<12><<<NP, 144, 0, st>>>(xa, W.x1w, W.x1b, xb, ACT_ELU);
    ein_k<12><<<NP, 144, 0, st>>>(xb, W.x2w, W.x2b, xa, ACT_NONE);
    fx_dw_k<12><<<NP, 256, 0, st>>>(xa, h1, fts, idxb, cpf, Cf, cin, dm, M, Pq, W.dww, dwo);
  } else {
    ein_k<16><<<NP, 256, 0, st>>>(xa, W.x1w, W.x1b, xb, ACT_ELU);
    ein_k<16><<<NP, 256, 0, st>>>(xb, W.x2w, W.x2b, xa, ACT_NONE);
    fx_dw_k<16><<<NP, 256, 0, st>>>(xa, h1, fts, idxb, cpf, Cf, cin, dm, M, Pq, W.dww, dwo);
  }
  dense(st, dwo, cin * dm, NP, cin * dm, W.pww, W.pwb, out, ldd, coff, C, ACT_ELU);
}

extern "C" void kernel_launch(void* const* d_in, const int* in_sizes, int n_in,
                              void* d_out, int out_size, void* d_ws, size_t ws_size,
                              hipStream_t stream)
{
  (void)in_sizes; (void)n_in; (void)out_size; (void)ws_size;
  auto F = [&](int i) { return (const float*)d_in[i]; };
  const float* points = F(109);  // (8, 2048, 6)

  // ---- workspace carving (256B aligned) ----
  char*  base = (char*)d_ws;
  size_t off  = 0;
  auto allocF = [&](size_t n) -> float* {
    float* p = (float*)(base + off); off += ((n * 4 + 255) / 256) * 256; return p;
  };
  auto allocI = [&](size_t n) -> int* {
    int* p = (int*)(base + off); off += ((n * 4 + 255) / 256) * 256; return p;
  };
  float* pts0 = allocF((size_t)8 * 2048 * 3);
  float* qrs1 = allocF((size_t)8 * 768 * 3);
  float* qrs2 = allocF((size_t)8 * 384 * 3);
  float* qrs3 = allocF((size_t)8 * 128 * 3);
  float* f0   = allocF((size_t)8 * 2048 * 64);
  float* f1   = allocF((size_t)8 * 2048 * 128);
  float* f2   = allocF((size_t)8 * 768 * 256);
  float* f3   = allocF((size_t)8 * 384 * 512);
  float* f4   = allocF((size_t)8 * 128 * 960);
  float* dA   = allocF((size_t)8 * 384 * 512);
  float* dB   = allocF((size_t)8 * 768 * 256);
  float* dC   = allocF((size_t)8 * 2048 * 128);
  float* fcA  = allocF((size_t)8 * 2048 * 128);
  float* fcB  = allocF((size_t)8 * 2048 * 128);
  int*   fidx = allocI((size_t)8 * 768);
  int*   idxb = allocI((size_t)8 * 2048 * 16);
  float* locb = allocF((size_t)8 * 2048 * 16 * 3);
  float* h0   = allocF((size_t)8 * 2048 * 12 * 64);
  float* h1   = allocF((size_t)8 * 2048 * 12 * 64);
  float* xa   = allocF((size_t)8 * 2048 * 16 * 16);
  float* xb   = allocF((size_t)8 * 2048 * 16 * 16);
  float* dwo  = allocF((size_t)8 * 2048 * 96 * 4);
  float* catb = allocF((size_t)8 * 2048 * 256);

  const int R = 8 * 2048;  // full-resolution row count

  // pts0 = points[:,:,:3];  f0 = elu(points[:,:,3:] @ dense1)
  copy_cols_k<<<(R * 3 + 255) / 256, 256, 0, stream>>>(points, 6, 0, pts0, 3, 0, R, 3);
  dense(stream, points + 3, 6, R, 3, F(1), F(0), f0, 64, 0, 64, ACT_ELU);

  // XConv 0: K=12 D=1 P=2048 C=128 (qrs == pts)
  XcP xc0 = mk_xc(d_in, 14, false);
  run_xconv(stream, xc0, pts0, 2048, f0, 64, pts0, 2048, 12, 1, 32, 4, 128,
            f1, 128, 0, idxb, locb, h0, h1, xa, xb, dwo);

  // FPS 2048 -> 768 ; XConv 1: K=16 D=1 C=256
  fps_k<<<8, 256, 0, stream>>>(pts0, 2048, 768, fidx);
  gather3_k<<<(8 * 768 + 255) / 256, 256, 0, stream>>>(pts0, fidx, 2048, 768, qrs1);
  XcP xc1 = mk_xc(d_in, 27, false);
  run_xconv(stream, xc1, pts0, 2048, f1, 128, qrs1, 768, 16, 1, 32, 2, 256,
            f2, 256, 0, idxb, locb, h0, h1, xa, xb, dwo);

  // FPS 768 -> 384 ; XConv 2: K=16 D=2 C=512
  fps_k<<<8, 256, 0, stream>>>(qrs1, 768, 384, fidx);
  gather3_k<<<(8 * 384 + 255) / 256, 256, 0, stream>>>(qrs1, fidx, 768, 384, qrs2);
  XcP xc2 = mk_xc(d_in, 40, false);
  run_xconv(stream, xc2, qrs1, 768, f2, 256, qrs2, 384, 16, 2, 64, 2, 512,
            f3, 512, 0, idxb, locb, h0, h1, xa, xb, dwo);

  // FPS 384 -> 128 ; XConv 3: K=16 D=2 C=768 + global branch (concat g first)
  fps_k<<<8, 256, 0, stream>>>(qrs2, 384, 128, fidx);
  gather3_k<<<(8 * 128 + 255) / 256, 256, 0, stream>>>(qrs2, fidx, 384, 128, qrs3);
  dense(stream, qrs3, 3, 8 * 128, 3, F(59), F(58), h0, 192, 0, 192, ACT_ELU);       // g0
  dense(stream, h0, 192, 8 * 128, 192, F(61), F(60), f4, 960, 0, 192, ACT_ELU);     // g1
  XcP xc3 = mk_xc(d_in, 53, true);
  run_xconv(stream, xc3, qrs2, 384, f3, 512, qrs3, 128, 16, 2, 128, 2, 768,
            f4, 960, 192, idxb, locb, h0, h1, xa, xb, dwo);

  // XDConv 0: pts=qrs3(128, f4:960ch) -> qrs2(384), C=512; fuse with f3
  XcP xd0 = mk_xc(d_in, 70, false);
  run_xconv(stream, xd0, qrs3, 128, f4, 960, qrs2, 384, 16, 2, 192, 1, 512,
            catb, 1024, 0, idxb, locb, h0, h1, xa, xb, dwo);
  copy_cols_k<<<((8 * 384 * 512) + 255) / 256, 256, 0, stream>>>(f3, 512, 0, catb, 1024, 512, 8 * 384, 512);
  dense(stream, catb, 1024, 8 * 384, 1024, F(7), F(6), dA, 512, 0, 512, ACT_ELU);

  // XDConv 1: pts=qrs2(384, dA:512ch) -> qrs1(768), C=256; fuse with f2
  XcP xd1 = mk_xc(d_in, 83, false);
  run_xconv(stream, xd1, qrs2, 384, dA, 512, qrs1, 768, 16, 1, 128, 1, 256,
            catb, 512, 0, idxb, locb, h0, h1, xa, xb, dwo);
  copy_cols_k<<<((8 * 768 * 256) + 255) / 256, 256, 0, stream>>>(f2, 256, 0, catb, 512, 256, 8 * 768, 256);
  dense(stream, catb, 512, 8 * 768, 512, F(9), F(8), dB, 256, 0, 256, ACT_ELU);

  // XDConv 2: pts=qrs1(768, dB:256ch) -> pts0(2048), C=128; fuse with f1
  XcP xd2 = mk_xc(d_in, 96, false);
  run_xconv(stream, xd2, qrs1, 768, dB, 256, pts0, 2048, 12, 1, 64, 1, 128,
            catb, 256, 0, idxb, locb, h0, h1, xa, xb, dwo);
  copy_cols_k<<<((R * 128) + 255) / 256, 256, 0, stream>>>(f1, 128, 0, catb, 256, 128, R, 128);
  dense(stream, catb, 256, R, 256, F(11), F(10), dC, 128, 0, 128, ACT_ELU);

  // FC stack + sigmoid head
  dense(stream, dC,  128, R, 128, F(3),  F(2),  fcA, 128, 0, 128, ACT_ELU);
  dense(stream, fcA, 128, R, 128, F(5),  F(4),  fcB, 128, 0, 128, ACT_ELU);
  dense(stream, fcB, 128, R, 128, F(13), F(12), (float*)d_out, 2, 0, 2, ACT_SIG);
}